// MultimodalTokenizerLoss_51531017617470
// MI455X (gfx1250) — compile-verified
//
#include <hip/hip_runtime.h>
#include <hip/hip_bf16.h>

typedef float v2f __attribute__((ext_vector_type(2)));
typedef float v8f __attribute__((ext_vector_type(8)));

#define NROW   8192
#define NEDGE  262144
#define NBLK   512
#define BLOCK  256
#define NTHREADS (NBLK * BLOCK)   // 131072 -> 2 iterations per thread
#define EPS_F  1e-8f

// ds_swizzle_b32 group-of-32 XOR patterns: offset = xor<<10 | or<<5 | and(0x1F)
template <int IMM>
__device__ __forceinline__ float swz(float x) {
    return __int_as_float(__builtin_amdgcn_ds_swizzle(__float_as_int(x), IMM));
}

// Full-precision wave32 reduction using CDNA5 WMMA:
//   D = A(ones 16x4) x B(4x16) + 0  =>  D[m][n] = B[0][n]+B[1][n]+B[2][n]+B[3][n]
// With B vgpr0 = s (per lane), vgpr1 = 0:
//   B[0][n] = s_n (lanes 0-15), B[2][n] = s_{n+16} (lanes 16-31)
// so every lane's D[0] element = s_{L%16} + s_{L%16+16}. Four ds_swizzle XOR
// steps finish 16 -> 1. EXEC is all-ones here (called unconditionally).
__device__ __forceinline__ float wave_reduce_wmma(float s) {
    v2f a; a.x = 1.0f; a.y = 1.0f;   // A = ones(16x4)
    v2f b; b.x = s;    b.y = 0.0f;   // B rows 0/2 carry lane partials
    v8f c = {};
    c = __builtin_amdgcn_wmma_f32_16x16x4_f32(
        /*neg_a=*/false, a, /*neg_b=*/false, b,
        /*c_mod=*/(short)0, c, /*reuse_a=*/false, /*reuse_b=*/false);
    float h = c[0];                  // s_{L%16} + s_{L%16+16}, dup'd in both halves
    h += swz<0x041F>(h);             // XOR 1
    h += swz<0x081F>(h);             // XOR 2
    h += swz<0x101F>(h);             // XOR 4
    h += swz<0x201F>(h);             // XOR 8
    return h;                        // wave total, uniform across lanes
}

__device__ __forceinline__ float block_reduce(float s, float* lds) {
    float w = wave_reduce_wmma(s);
    const int lane = threadIdx.x & 31;
    const int wid  = threadIdx.x >> 5;
    if (lane == 0) lds[wid] = w;
    __syncthreads();
    float t = 0.0f;
    if (threadIdx.x == 0) {
        #pragma unroll
        for (int i = 0; i < BLOCK / 32; ++i) t += lds[i];  // fixed order: deterministic
    }
    return t;
}

// Pass 1: gather + log, one partial sum per block into d_ws.
__global__ void __launch_bounds__(BLOCK)
gather_loss_kernel(const float* __restrict__ adj,
                   const int*   __restrict__ src,   // edge_index row 0
                   const int*   __restrict__ dst,   // edge_index row 1
                   const int2*  __restrict__ neg,   // neg_edges, (row, col) pairs
                   float*       __restrict__ block_partials) {
    __shared__ float lds[BLOCK / 32];
    const int tid = blockIdx.x * BLOCK + threadIdx.x;

    float s = 0.0f;
    #pragma unroll
    for (int k = 0; k < NEDGE / NTHREADS; ++k) {
        const int e = tid + k * NTHREADS;
        // Index loads first; the unrolled body lets the compiler batch all
        // four independent adjacency gathers under one loadcnt wait.
        const int  r0 = src[e];
        const int  c0 = dst[e];
        const int2 nc = neg[e];
        const float p = adj[(size_t)r0   * NROW + (size_t)c0  ];
        const float q = adj[(size_t)nc.x * NROW + (size_t)nc.y];
        s += __logf(p + EPS_F);
        s += __logf(1.0f - q + EPS_F);
    }

    const float t = block_reduce(s, lds);
    if (threadIdx.x == 0) block_partials[blockIdx.x] = t;
}

// Pass 2: deterministic tree over the 512 block partials + codebook terms.
__global__ void __launch_bounds__(BLOCK)
finalize_kernel(const float* __restrict__ block_partials,
                const float* __restrict__ codebook,
                float*       __restrict__ out) {
    __shared__ float lds[BLOCK / 32];
    const int tid = threadIdx.x;
    float s = block_partials[tid] + block_partials[tid + BLOCK];  // NBLK = 2*BLOCK
    const float t = block_reduce(s, lds);
    if (tid == 0) {
        const float rec_loss = -t * (1.0f / (float)NEDGE);  // pos_loss + neg_loss
        out[0] = rec_loss + codebook[0] + codebook[1] + codebook[2] + codebook[3];
    }
}

extern "C" void kernel_launch(void* const* d_in, const int* in_sizes, int n_in,
                              void* d_out, int out_size, void* d_ws, size_t ws_size,
                              hipStream_t stream) {
    const float* adj      = (const float*)d_in[0];
    const float* codebook = (const float*)d_in[1];
    const int*   edge     = (const int*)d_in[2];   // [2, E] row-major
    const int2*  neg      = (const int2*)d_in[3];  // [E, 2] -> int2 pairs
    float*       out      = (float*)d_out;
    float*       partials = (float*)d_ws;          // NBLK floats of scratch

    const int* src = edge;
    const int* dst = edge + NEDGE;

    gather_loss_kernel<<<NBLK, BLOCK, 0, stream>>>(adj, src, dst, neg, partials);
    finalize_kernel<<<1, BLOCK, 0, stream>>>(partials, codebook, out);
}